// MOERouter_34351148433721
// MI455X (gfx1250) — compile-verified
//
#include <hip/hip_runtime.h>

typedef __attribute__((ext_vector_type(2))) float v2f;
typedef __attribute__((ext_vector_type(8))) float v8f;

#define HIDDEN  4096
#define EXPERTS 64
#define TOPK    8
#define TOKENS  8192

#define BK        64            // K-chunk staged in LDS
#define LDSTRIDE  (BK + 4)      // padded row stride (floats): bank-conflict-free b64 reads

// ---------------------------------------------------------------------------
// Kernel 1: router_logits = x @ W^T + b via V_WMMA_F32_16X16X4_F32 (fp32 WMMA
// keeps exact top-k ordering vs the fp32 reference; bf16 would risk tie flips).
// Block: 256 threads (8 waves). Block tile: 64 tokens x 64 experts.
// Wave w: token subgroup tg = w>>1 (16 tokens), expert subgroup eg = w&1 (32 experts).
// ---------------------------------------------------------------------------
__global__ __launch_bounds__(256) void moe_gemm_logits(
    const float* __restrict__ x, const float* __restrict__ W,
    const float* __restrict__ bias, float* __restrict__ logits)
{
  __shared__ __align__(16) float ldsX[64 * LDSTRIDE];
  __shared__ __align__(16) float ldsW[64 * LDSTRIDE];

  const int tid     = threadIdx.x;
  const int lane    = tid & 31;
  const int wave    = tid >> 5;
  const int tg      = wave >> 1;   // 0..3  (16-token group)
  const int eg      = wave & 1;    // 0..1  (32-expert group)
  const int tokBase = blockIdx.x * 64;

  // WMMA fp32 fragment addressing (ISA 7.12.2, 32-bit 16x4 A / 4x16 B):
  //   lanes 0-15 hold K-pair {0,1}, lanes 16-31 hold K-pair {2,3}; m/n = lane&15
  const int m  = lane & 15;
  const int kb = (lane >> 4) * 2;

  v8f acc0 = {};   // experts [eg*32 +  0 .. +15]
  v8f acc1 = {};   // experts [eg*32 + 16 .. +31]

  for (int kc = 0; kc < HIDDEN; kc += BK) {
    __syncthreads();
    // Stage x[tokBase..+63][kc..+63] and W[0..63][kc..+63] -> LDS, coalesced float4.
#pragma unroll
    for (int i = 0; i < 4; ++i) {
      const int idx = tid + 256 * i;        // 1024 float4 slots per tile
      const int row = idx >> 4;
      const int col = (idx & 15) << 2;
      const float4 xv = *(const float4*)(x + (size_t)(tokBase + row) * HIDDEN + kc + col);
      *(float4*)(ldsX + row * LDSTRIDE + col) = xv;
      const float4 wv = *(const float4*)(W + (size_t)row * HIDDEN + kc + col);
      *(float4*)(ldsW + row * LDSTRIDE + col) = wv;
    }
    __syncthreads();

    // Warm the next x chunk (global_prefetch_b8): 128 cachelines = 64 rows x 2 lines.
    if (kc + BK < HIDDEN && tid < 128) {
      const int row = tid >> 1;
      const int col = (tid & 1) * 32;
      __builtin_prefetch(x + (size_t)(tokBase + row) * HIDDEN + kc + BK + col, 0, 3);
    }

#pragma unroll
    for (int ks = 0; ks < BK / 4; ++ks) {
      const int kk = ks * 4;
      // A fragment: a.x = x[m][kk+kb], a.y = x[m][kk+kb+1]
      const float* ax = ldsX + (tg * 16 + m) * LDSTRIDE + kk + kb;
      v2f a; a.x = ax[0]; a.y = ax[1];
      // B fragment: b.x = B[kb][n] = W[n][kk+kb] (row-major W tile -> contiguous pair)
      const float* bw0 = ldsW + (eg * 32 + m) * LDSTRIDE + kk + kb;
      v2f b0; b0.x = bw0[0]; b0.y = bw0[1];
      acc0 = __builtin_amdgcn_wmma_f32_16x16x4_f32(false, a, false, b0,
                                                   (short)0, acc0, false, false);
      const float* bw1 = bw0 + 16 * LDSTRIDE;
      v2f b1; b1.x = bw1[0]; b1.y = bw1[1];
      acc1 = __builtin_amdgcn_wmma_f32_16x16x4_f32(false, a, false, b1,
                                                   (short)0, acc1, false, false);
    }
  }

  // D layout: VGPR r -> row (lane>>4)*8 + r, col = lane&15.
  const int token = tokBase + tg * 16 + (lane >> 4) * 8;
  const int e0    = eg * 32 + m;
  const float bia0 = bias[e0];
  const float bia1 = bias[e0 + 16];
#pragma unroll
  for (int r = 0; r < 8; ++r) {
    logits[(size_t)(token + r) * EXPERTS + e0]      = acc0[r] + bia0;
    logits[(size_t)(token + r) * EXPERTS + e0 + 16] = acc1[r] + bia1;
  }
}

// ---------------------------------------------------------------------------
// Kernel 2: per-token top-8 + renormalized softmax.
// renorm(softmax(all)[top8]) == softmax(top8 logits), so only 8 exps needed.
// Strict '>' argmax sweeps == jax.lax.top_k (descending, lowest index on tie).
// All array indices static after unroll -> stays in VGPRs (no scratch).
// ---------------------------------------------------------------------------
__global__ __launch_bounds__(256) void moe_topk(
    const float* __restrict__ logits,
    float* __restrict__ probs_out, int* __restrict__ idx_out)
{
  const int t = blockIdx.x * 256 + threadIdx.x;
  float v[EXPERTS];
#pragma unroll
  for (int i = 0; i < EXPERTS / 4; ++i) {
    const float4 l = *(const float4*)(logits + (size_t)t * EXPERTS + i * 4);
    v[4 * i + 0] = l.x; v[4 * i + 1] = l.y;
    v[4 * i + 2] = l.z; v[4 * i + 3] = l.w;
  }

  float topv[TOPK];
  int   topi[TOPK];
#pragma unroll
  for (int k = 0; k < TOPK; ++k) {
    float best = -__builtin_inff();
    int   bi   = 0;
#pragma unroll
    for (int e = 0; e < EXPERTS; ++e) {
      if (v[e] > best) { best = v[e]; bi = e; }
    }
    topv[k] = best;
    topi[k] = bi;
    // predicated clear (static indices -> no scratch spill)
#pragma unroll
    for (int e = 0; e < EXPERTS; ++e) {
      v[e] = (e == bi) ? -__builtin_inff() : v[e];
    }
  }

  const float mx = topv[0];           // top-1 == global max
  float p[TOPK];
  float s = 0.f;
#pragma unroll
  for (int k = 0; k < TOPK; ++k) { p[k] = __expf(topv[k] - mx); s += p[k]; }
  const float inv = 1.f / s;
#pragma unroll
  for (int k = 0; k < TOPK; ++k) {
    probs_out[(size_t)t * TOPK + k] = p[k] * inv;
    idx_out[(size_t)t * TOPK + k]   = topi[k];
  }
}

// ---------------------------------------------------------------------------
// Kernel 3: expert_mask[T, E, K] = (top_k_indices[t, k] == e), fully coalesced.
// ---------------------------------------------------------------------------
__global__ __launch_bounds__(256) void moe_mask(
    const int* __restrict__ idx, int* __restrict__ mask)
{
  const size_t i = (size_t)blockIdx.x * 256 + threadIdx.x;  // t*512 + e*8 + k
  const int    k = (int)(i & 7);
  const int    e = (int)((i >> 3) & 63);
  const size_t t = i >> 9;
  mask[i] = (idx[t * TOPK + k] == e) ? 1 : 0;
}

extern "C" void kernel_launch(void* const* d_in, const int* in_sizes, int n_in,
                              void* d_out, int out_size, void* d_ws, size_t ws_size,
                              hipStream_t stream) {
  (void)in_sizes; (void)n_in; (void)out_size; (void)d_ws; (void)ws_size;
  const float* x = (const float*)d_in[0];   // [TOKENS, HIDDEN]
  const float* W = (const float*)d_in[1];   // [EXPERTS, HIDDEN]
  const float* b = (const float*)d_in[2];   // [EXPERTS]

  float* out    = (float*)d_out;
  float* logits = out;                                        // T*E f32
  float* probs  = out + (size_t)TOKENS * EXPERTS;             // T*K f32
  int*   idx    = (int*)(probs + (size_t)TOKENS * TOPK);      // T*K i32 (bit pattern)
  int*   mask   = (int*)(idx + (size_t)TOKENS * TOPK);        // T*E*K i32

  moe_gemm_logits<<<TOKENS / 64, 256, 0, stream>>>(x, W, b, logits);
  moe_topk<<<TOKENS / 256, 256, 0, stream>>>(logits, probs, idx);
  moe_mask<<<(TOKENS * EXPERTS * TOPK) / 256, 256, 0, stream>>>(idx, mask);
}